// GCN_73967926772366
// MI455X (gfx1250) — compile-verified
//
#include <hip/hip_runtime.h>

typedef __attribute__((ext_vector_type(2))) float v2f;
typedef __attribute__((ext_vector_type(8))) float v8f;

// ---------------------------------------------------------------------------
// Degree / normalization kernels
// ---------------------------------------------------------------------------
__global__ void gcn_init_deg(float* __restrict__ deg, int n) {
    int i = blockIdx.x * blockDim.x + threadIdx.x;
    if (i < n) deg[i] = 1.0f;              // self-loop contributes 1
}

__global__ void gcn_count_deg(const long long* __restrict__ cols, float* __restrict__ deg,
                              long long E) {
    long long e = (long long)blockIdx.x * blockDim.x + threadIdx.x;
    if (e < E) atomicAdd(&deg[cols[e]], 1.0f);
}

__global__ void gcn_dinv(float* __restrict__ deg, int n) {
    int i = blockIdx.x * blockDim.x + threadIdx.x;
    if (i < n) {
        float d = deg[i];
        deg[i] = (d > 0.0f) ? rsqrtf(d) : 0.0f;
    }
}

// ---------------------------------------------------------------------------
// WMMA f32 GEMM: D[nrows x NCOL] = A[nrows x K] @ B[K x NCOL]
// One wave32 computes one 16x16 tile using V_WMMA_F32_16X16X4_F32.
// A frag (16x4): lanes 0-15 hold M=lane,K=k0+{0,1}; lanes 16-31 M=lane-16,K=k0+{2,3}
// B frag (4x16): v0: K=k0(+2 for hi lanes), N=lane&15 ; v1: K=k0+1(+2)
// C/D (16x16):   vgpr v -> M = v + 8*(lane>=16), N = lane&15
// ---------------------------------------------------------------------------
template <int K, int NCOL>
__global__ void gcn_wmma_gemm(const float* __restrict__ A, const float* __restrict__ B,
                              float* __restrict__ D, int nrows) {
    constexpr int COL_TILES  = (NCOL + 15) / 16;
    constexpr bool TAIL      = (NCOL % 16) != 0;

    const int wave    = (int)((blockIdx.x * (long long)blockDim.x + threadIdx.x) >> 5);
    const int lane    = threadIdx.x & 31;
    const int rowTile = wave / COL_TILES;
    const int colTile = wave % COL_TILES;
    if (rowTile * 16 >= nrows) return;      // wave-uniform exit keeps EXEC full

    const int m    = rowTile * 16 + (lane & 15);   // A row for this lane
    const int kOff = (lane >> 4) * 2;              // 0 or 2
    const int nB   = colTile * 16 + (lane & 15);   // B column for this lane
    const bool nOOB = TAIL && (nB >= NCOL);

    v8f c = {0.f, 0.f, 0.f, 0.f, 0.f, 0.f, 0.f, 0.f};

    #pragma unroll 4
    for (int k0 = 0; k0 < K; k0 += 4) {
        v2f a;
        a.x = A[(long long)m * K + k0 + kOff];
        a.y = A[(long long)m * K + k0 + kOff + 1];
        v2f b;
        b.x = nOOB ? 0.0f : B[(k0 + kOff) * NCOL + nB];
        b.y = nOOB ? 0.0f : B[(k0 + kOff + 1) * NCOL + nB];
        c = __builtin_amdgcn_wmma_f32_16x16x4_f32(false, a, false, b,
                                                  (short)0, c, false, false);
    }

    const int mBase = rowTile * 16 + (lane >> 4) * 8;
    if (!nOOB) {
        #pragma unroll
        for (int v = 0; v < 8; ++v)
            D[(long long)(mBase + v) * NCOL + nB] = c[v];
    }
}

// ---------------------------------------------------------------------------
// Edge aggregation: agg[col] += hl[row] * dinv[row]*dinv[col]
// Edges e in [0,E) come from edge_idx; e in [E, E+N) are self-loops.
// One thread handles one float4 chunk of one edge; agg stays L2-resident.
// ---------------------------------------------------------------------------
template <int F>
__global__ void gcn_aggregate(const float* __restrict__ hl,
                              const long long* __restrict__ rows,
                              const long long* __restrict__ cols,
                              const float* __restrict__ dinv,
                              float* __restrict__ agg,
                              long long E, int n) {
    constexpr int FV = F / 4;
    long long idx = (long long)blockIdx.x * blockDim.x + threadIdx.x;
    long long total = (E + (long long)n) * (long long)FV;
    if (idx >= total) return;
    long long e = idx / FV;
    int f = (int)(idx - e * FV) * 4;

    long long r, cgt;
    if (e < E) { r = rows[e]; cgt = cols[e]; }
    else       { r = cgt = e - E; }          // self loop

    float nrm = dinv[r] * dinv[cgt];
    const float4 h4 = *(const float4*)(hl + r * (long long)F + f);
    float* dst = agg + cgt * (long long)F + f;
    atomicAdd(dst + 0, h4.x * nrm);
    atomicAdd(dst + 1, h4.y * nrm);
    atomicAdd(dst + 2, h4.z * nrm);
    atomicAdd(dst + 3, h4.w * nrm);
}

// ---------------------------------------------------------------------------
// Elementwise epilogues
// ---------------------------------------------------------------------------
__global__ void gcn_bias_bn_relu(const float* __restrict__ agg,
                                 const float* __restrict__ b, const float* __restrict__ g,
                                 const float* __restrict__ be, const float* __restrict__ m,
                                 const float* __restrict__ v,
                                 float* __restrict__ out, long long total, int F) {
    long long i = (long long)blockIdx.x * blockDim.x + threadIdx.x;
    if (i >= total) return;
    int f = (int)(i % F);
    float h = agg[i] + b[f];
    h = g[f] * (h - m[f]) * rsqrtf(v[f] + 1e-5f) + be[f];
    out[i] = fmaxf(h, 0.0f);
}

__global__ void gcn_bias_add(float* __restrict__ out, const float* __restrict__ b,
                             long long total, int F) {
    long long i = (long long)blockIdx.x * blockDim.x + threadIdx.x;
    if (i >= total) return;
    out[i] = out[i] + b[(int)(i % F)];
}

// ---------------------------------------------------------------------------
static inline unsigned div_up_ll(long long a, long long b) {
    return (unsigned)((a + b - 1) / b);
}

extern "C" void kernel_launch(void* const* d_in, const int* in_sizes, int n_in,
                              void* d_out, int out_size, void* d_ws, size_t ws_size,
                              hipStream_t stream) {
    constexpr int FIN = 128, H = 64, C = 40;

    const float*     x    = (const float*)d_in[0];
    const long long* eidx = (const long long*)d_in[1];
    const float* W1  = (const float*)d_in[2];
    const float* b1  = (const float*)d_in[3];
    const float* g1  = (const float*)d_in[4];
    const float* be1 = (const float*)d_in[5];
    const float* m1  = (const float*)d_in[6];
    const float* v1  = (const float*)d_in[7];
    const float* W2  = (const float*)d_in[8];
    const float* b2  = (const float*)d_in[9];
    const float* g2  = (const float*)d_in[10];
    const float* be2 = (const float*)d_in[11];
    const float* m2  = (const float*)d_in[12];
    const float* v2  = (const float*)d_in[13];
    const float* W3  = (const float*)d_in[14];
    const float* b3  = (const float*)d_in[15];

    const int n = in_sizes[0] / FIN;          // 100000
    const long long E = in_sizes[1] / 2;      // 1.6M
    const long long* rows = eidx;             // edge_idx[0]
    const long long* cols = eidx + E;         // edge_idx[1]

    // Workspace layout (256B aligned slices)
    char* ws = (char*)d_ws;
    auto align256 = [](size_t s) { return (s + 255) & ~(size_t)255; };
    size_t off = 0;
    float* dinv = (float*)(ws + off); off += align256((size_t)n * sizeof(float));
    float* bufA = (float*)(ws + off); off += align256((size_t)n * H * sizeof(float));
    float* bufB = (float*)(ws + off); off += align256((size_t)n * H * sizeof(float));
    float* bufC = (float*)(ws + off);          // n * C floats
    float* out  = (float*)d_out;

    const int TB = 256;

    // --- symmetric gcn_norm: deg (with self loops) -> dinv ---
    gcn_init_deg<<<div_up_ll(n, TB), TB, 0, stream>>>(dinv, n);
    gcn_count_deg<<<div_up_ll(E, TB), TB, 0, stream>>>(cols, dinv, E);
    gcn_dinv<<<div_up_ll(n, TB), TB, 0, stream>>>(dinv, n);

    const int rowTiles = (n + 15) / 16;

    // --- Layer 1: lin = x @ W1 ; agg ; bias+bn+relu -> bufA ---
    {
        long long waves = (long long)rowTiles * ((H + 15) / 16);
        gcn_wmma_gemm<FIN, H><<<div_up_ll(waves * 32, TB), TB, 0, stream>>>(x, W1, bufA, n);
        hipMemsetAsync(bufB, 0, (size_t)n * H * sizeof(float), stream);
        long long tot = (E + n) * (long long)(H / 4);
        gcn_aggregate<H><<<div_up_ll(tot, TB), TB, 0, stream>>>(bufA, rows, cols, dinv, bufB, E, n);
        long long te = (long long)n * H;
        gcn_bias_bn_relu<<<div_up_ll(te, TB), TB, 0, stream>>>(bufB, b1, g1, be1, m1, v1, bufA, te, H);
    }

    // --- Layer 2: lin = h1 @ W2 ; agg ; bias+bn+relu -> bufB ---
    {
        long long waves = (long long)rowTiles * ((H + 15) / 16);
        gcn_wmma_gemm<H, H><<<div_up_ll(waves * 32, TB), TB, 0, stream>>>(bufA, W2, bufB, n);
        hipMemsetAsync(bufA, 0, (size_t)n * H * sizeof(float), stream);
        long long tot = (E + n) * (long long)(H / 4);
        gcn_aggregate<H><<<div_up_ll(tot, TB), TB, 0, stream>>>(bufB, rows, cols, dinv, bufA, E, n);
        long long te = (long long)n * H;
        gcn_bias_bn_relu<<<div_up_ll(te, TB), TB, 0, stream>>>(bufA, b2, g2, be2, m2, v2, bufB, te, H);
    }

    // --- Layer 3: lin = h2 @ W3 ; agg into d_out ; + b3 ---
    {
        long long waves = (long long)rowTiles * ((C + 15) / 16);
        gcn_wmma_gemm<H, C><<<div_up_ll(waves * 32, TB), TB, 0, stream>>>(bufB, W3, bufC, n);
        hipMemsetAsync(out, 0, (size_t)n * C * sizeof(float), stream);
        long long tot = (E + n) * (long long)(C / 4);
        gcn_aggregate<C><<<div_up_ll(tot, TB), TB, 0, stream>>>(bufC, rows, cols, dinv, out, E, n);
        long long te = (long long)n * C;
        gcn_bias_add<<<div_up_ll(te, TB), TB, 0, stream>>>(out, b3, te, C);
    }
}